// TokenRouter_91018946937084
// MI455X (gfx1250) — compile-verified
//
#include <hip/hip_runtime.h>
#include <math.h>

#define HIDDEN      2048
#define BOTTLENECK  64
#define NUM_EXPERTS 8
#define LN_EPS      1e-5f
#define TOKENS_PER_WG   128
#define TOKENS_PER_WAVE 32

typedef float v2f __attribute__((ext_vector_type(2)));
typedef float v8f __attribute__((ext_vector_type(8)));

// ---------------------------------------------------------------------------
// Kernel 1: fold LayerNorm affine params into the down-projection.
//   wp[d][h]  = gamma[h] * w_down[d][h]
//   sbuf[d]   = sum_h gamma[h] * w_down[d][h]   (for the -mu term)
//   bbuf[d]   = sum_h beta[h]  * w_down[d][h]   (beta contribution)
// ---------------------------------------------------------------------------
__global__ void __launch_bounds__(256)
router_prep(const float* __restrict__ w_down,
            const float* __restrict__ gamma,
            const float* __restrict__ beta,
            float* __restrict__ wp,
            float* __restrict__ sbuf,
            float* __restrict__ bbuf) {
  const int d   = blockIdx.x;
  const int tid = threadIdx.x;
  float sg = 0.f, sb = 0.f;
  for (int h = tid; h < HIDDEN; h += 256) {
    float w = w_down[d * HIDDEN + h];
    float g = gamma[h];
    wp[d * HIDDEN + h] = g * w;
    sg += g * w;
    sb += beta[h] * w;
  }
  __shared__ float rs[256], rb[256];
  rs[tid] = sg; rb[tid] = sb;
  __syncthreads();
  for (int s = 128; s > 0; s >>= 1) {
    if (tid < s) { rs[tid] += rs[tid + s]; rb[tid] += rb[tid + s]; }
    __syncthreads();
  }
  if (tid == 0) { sbuf[d] = rs[0]; bbuf[d] = rb[0]; }
}

// ---------------------------------------------------------------------------
// Kernel 2: fused LN-stats + bottleneck GEMM (f32 WMMA) + silu + router
//           matvec + top-2 softmax. One wave owns 32 tokens x 64 dims.
// ---------------------------------------------------------------------------
__global__ void __launch_bounds__(128)
router_main(const float* __restrict__ x,
            const float* __restrict__ wp,
            const float* __restrict__ sbuf,
            const float* __restrict__ bbuf,
            const float* __restrict__ w_up,
            float* __restrict__ out) {
  __shared__ float  zbuf[TOKENS_PER_WG * 65];          // padded: stride 65
  __shared__ float2 stats[TOKENS_PER_WG];              // (mu, rsqrt)
  __shared__ float  wup_s[NUM_EXPERTS * BOTTLENECK];
  __shared__ float2 sb_s[BOTTLENECK];                  // (s_d, b_d)

  const int tid  = threadIdx.x;
  const int wave = tid >> 5;
  const int lane = tid & 31;
  const int col  = lane & 15;   // matrix row (A) / column (B,C,D)
  const int kh   = lane >> 4;   // which K-half of the 16x4 fragment

  for (int i = tid; i < NUM_EXPERTS * BOTTLENECK; i += 128) wup_s[i] = w_up[i];
  if (tid < BOTTLENECK) sb_s[tid] = make_float2(sbuf[tid], bbuf[tid]);
  __syncthreads();

  const int tok_wg = blockIdx.x * TOKENS_PER_WG;
  const int tok0   = tok_wg + wave * TOKENS_PER_WAVE + col;  // frag0 token row
  const int tok1   = tok0 + 16;                              // frag1 token row

  // A-fragment streams: lane l<16 reads K={k,k+1}, lane l>=16 K={k+2,k+3}
  const float* xr0 = x + (size_t)tok0 * HIDDEN + 2 * kh;
  const float* xr1 = x + (size_t)tok1 * HIDDEN + 2 * kh;
  // B-fragment base: row d = 16*n + col, same K-half striping as A
  const float* wpL = wp + (size_t)col * HIDDEN + 2 * kh;

  v8f acc0[4], acc1[4];
  {
    v8f z = {};
    #pragma unroll
    for (int n = 0; n < 4; ++n) { acc0[n] = z; acc1[n] = z; }
  }
  float s1_0 = 0.f, s2_0 = 0.f, s1_1 = 0.f, s2_1 = 0.f;

  for (int k = 0; k < HIDDEN; k += 4) {
    v2f a0 = *(const v2f*)(xr0 + k);
    v2f a1 = *(const v2f*)(xr1 + k);
    // LayerNorm statistics ride along for free on the A stream
    s1_0 += a0.x + a0.y;  s2_0 += a0.x * a0.x + a0.y * a0.y;
    s1_1 += a1.x + a1.y;  s2_1 += a1.x * a1.x + a1.y * a1.y;
    #pragma unroll
    for (int n = 0; n < 4; ++n) {
      v2f b = *(const v2f*)(wpL + (size_t)(16 * n) * HIDDEN + k);
      acc0[n] = __builtin_amdgcn_wmma_f32_16x16x4_f32(
          false, a0, false, b, (short)0, acc0[n], false, false);
      acc1[n] = __builtin_amdgcn_wmma_f32_16x16x4_f32(
          false, a1, false, b, (short)0, acc1[n], false, false);
    }
  }

  // finish per-token LN stats: combine the two K-halves (wave32 -> xor 16)
  float t1_0 = s1_0 + __shfl_xor(s1_0, 16);
  float t2_0 = s2_0 + __shfl_xor(s2_0, 16);
  float t1_1 = s1_1 + __shfl_xor(s1_1, 16);
  float t2_1 = s2_1 + __shfl_xor(s2_1, 16);
  if (kh == 0) {
    const float inv = 1.f / (float)HIDDEN;
    float mu0 = t1_0 * inv;
    float r0  = rsqrtf(t2_0 * inv - mu0 * mu0 + LN_EPS);
    stats[wave * 32 + col] = make_float2(mu0, r0);
    float mu1 = t1_1 * inv;
    float r1  = rsqrtf(t2_1 * inv - mu1 * mu1 + LN_EPS);
    stats[wave * 32 + 16 + col] = make_float2(mu1, r1);
  }

  // epilogue: z = silu(r*(y - mu*s_d) + b_d), stage into padded LDS
  #pragma unroll
  for (int f = 0; f < 2; ++f) {
    #pragma unroll
    for (int n = 0; n < 4; ++n) {
      v8f a = f ? acc1[n] : acc0[n];
      const int d = 16 * n + col;
      const float2 sbv = sb_s[d];
      #pragma unroll
      for (int i = 0; i < 8; ++i) {
        const int row = i + 8 * kh;                 // C layout: VGPR i -> M=i / i+8
        const int tl  = wave * 32 + f * 16 + row;   // local token id
        const float2 st = stats[tl];
        float zp = st.y * (a[i] - st.x * sbv.x) + sbv.y;
        float zz = zp / (1.f + __expf(-zp));        // silu
        zbuf[tl * 65 + d] = zz;
      }
    }
  }
  __syncthreads();

  // phase 2: one thread per token -> 64x8 matvec + top-2 softmax
  {
    const int tl = tid;                 // 128 threads <-> 128 tokens
    float lg[NUM_EXPERTS];
    #pragma unroll
    for (int e = 0; e < NUM_EXPERTS; ++e) lg[e] = 0.f;
    for (int d = 0; d < BOTTLENECK; ++d) {
      float zv = zbuf[tl * 65 + d];     // stride 65 -> conflict-free
      #pragma unroll
      for (int e = 0; e < NUM_EXPERTS; ++e)
        lg[e] += zv * wup_s[e * BOTTLENECK + d];
    }
    float v1 = -3.4e38f, v2 = -3.4e38f;
    int   i1 = 0, i2 = 0;
    #pragma unroll
    for (int e = 0; e < NUM_EXPERTS; ++e) {
      float v = lg[e];
      if (v > v1)      { v2 = v1; i2 = i1; v1 = v; i1 = e; }
      else if (v > v2) { v2 = v;  i2 = e; }
    }
    float e2  = __expf(v2 - v1);
    float den = 1.f + e2;
    float w1  = 1.f / den, w2 = e2 / den;
    float* op = out + (size_t)(tok_wg + tl) * NUM_EXPERTS;
    #pragma unroll
    for (int e = 0; e < NUM_EXPERTS; ++e)
      op[e] = (e == i1) ? w1 : ((e == i2) ? w2 : 0.f);
  }
}

// ---------------------------------------------------------------------------
extern "C" void kernel_launch(void* const* d_in, const int* in_sizes, int n_in,
                              void* d_out, int out_size, void* d_ws, size_t ws_size,
                              hipStream_t stream) {
  (void)n_in; (void)out_size; (void)ws_size;
  const float* x      = (const float*)d_in[0];   // (4,4096,2048)
  const float* w_down = (const float*)d_in[1];   // (64,2048)
  const float* w_up   = (const float*)d_in[2];   // (8,64)
  const float* gamma  = (const float*)d_in[3];   // (2048)
  const float* beta   = (const float*)d_in[4];   // (2048)
  float*       out    = (float*)d_out;           // (4,4096,8)

  float* wp   = (float*)d_ws;                    // 64*2048 floats
  float* sbuf = wp + BOTTLENECK * HIDDEN;        // 64 floats
  float* bbuf = sbuf + BOTTLENECK;               // 64 floats

  const int n_tokens = in_sizes[0] / HIDDEN;     // 16384

  router_prep<<<BOTTLENECK, 256, 0, stream>>>(w_down, gamma, beta, wp, sbuf, bbuf);
  router_main<<<n_tokens / TOKENS_PER_WG, 128, 0, stream>>>(
      x, wp, sbuf, bbuf, w_up, out);
}